// RandLANet_43095701848395
// MI455X (gfx1250) — compile-verified
//
#include <hip/hip_runtime.h>
#include <hip/hip_bf16.h>
#include <math.h>

// ---------------------------------------------------------------------------
// RandLA-Net forward for MI455X (gfx1250).
// All matmuls go through V_WMMA_F32_16X16X4_F32 (full fp32 = reference
// numerics). Fast GEMM path: branch-free inner loop (clamped rows/cols,
// uniform concat split), 32x16 tile per wave (2 accumulators), float2 A loads.
// Gathers are float4-vectorized. Concats virtualized via two-pointer GEMMs;
// residual + BN + leakyReLU fused in GEMM epilogue; rel_pos+bb_mlp1 fused.
// ---------------------------------------------------------------------------

typedef float v2f __attribute__((ext_vector_type(2)));
typedef float v8f __attribute__((ext_vector_type(8)));

#define NEG_SLOPE 0.2f
#define WPB 8  // waves per block (256 threads, wave32)

// ---------------------------------------------------------------------------
// Fast WMMA GEMM: Y[M,Cout] = epi( concat(X1,X2)[M,C1+C2] @ W[C1+C2,Cout] )
// Requires C1 % 4 == 0 and C2 % 4 == 0 (C2 may be 0). One wave computes a
// 32x16 tile: rows [mt*32, mt*32+32) x cols [nt*16, nt*16+16).
// A lanes: 0-15 hold K=k0,k0+1 ; 16-31 hold K=k0+2,k0+3 (fp32 16x4 layout).
// ---------------------------------------------------------------------------
__global__ __launch_bounds__(256) void wgemm_fast_kernel(
    const float* __restrict__ X1, int C1,
    const float* __restrict__ X2, int C2,
    const float* __restrict__ W, const float* __restrict__ bias,
    const float* __restrict__ g, const float* __restrict__ bt,
    const float* __restrict__ R, float* __restrict__ Y,
    int M, int Cout, int act)
{
    const int wave = threadIdx.x >> 5;
    const int lane = threadIdx.x & 31;
    const int ntiles = (Cout + 15) >> 4;
    const long long mtiles = (M + 31) >> 5;
    const long long tile = (long long)blockIdx.x * WPB + wave;
    if (tile >= mtiles * ntiles) return;          // wave-uniform exit
    const int nt = (int)(tile % ntiles);
    const long long mt = tile / ntiles;
    const int half = lane >> 4;
    const int l16  = lane & 15;
    const int row0 = (int)(mt * 32) + l16;
    const int row1 = row0 + 16;
    const int r0c  = min(row0, M - 1);            // clamped: loads always legal
    const int r1c  = min(row1, M - 1);
    const int col  = nt * 16 + l16;
    const int colc = min(col, Cout - 1);

    v8f acc0 = {0.f,0.f,0.f,0.f,0.f,0.f,0.f,0.f};
    v8f acc1 = {0.f,0.f,0.f,0.f,0.f,0.f,0.f,0.f};
    const float* pb = W + (size_t)(half * 2) * Cout + colc;

    {   // section 1: columns [0, C1) from X1
        const float* pa0 = X1 + (size_t)r0c * C1 + half * 2;
        const float* pa1 = X1 + (size_t)r1c * C1 + half * 2;
        #pragma unroll 2
        for (int k0 = 0; k0 < C1; k0 += 4) {
            const v2f a0 = *(const v2f*)(pa0 + k0);
            const v2f a1 = *(const v2f*)(pa1 + k0);
            v2f b;
            b.x = pb[(size_t)k0 * Cout];
            b.y = pb[(size_t)k0 * Cout + Cout];
            acc0 = __builtin_amdgcn_wmma_f32_16x16x4_f32(false, a0, false, b, (short)0, acc0, false, false);
            acc1 = __builtin_amdgcn_wmma_f32_16x16x4_f32(false, a1, false, b, (short)0, acc1, false, false);
        }
        pb += (size_t)C1 * Cout;
    }
    if (C2 > 0) {  // section 2: columns [C1, C1+C2) from X2 (virtual concat)
        const float* pa0 = X2 + (size_t)r0c * C2 + half * 2;
        const float* pa1 = X2 + (size_t)r1c * C2 + half * 2;
        #pragma unroll 2
        for (int k0 = 0; k0 < C2; k0 += 4) {
            const v2f a0 = *(const v2f*)(pa0 + k0);
            const v2f a1 = *(const v2f*)(pa1 + k0);
            v2f b;
            b.x = pb[(size_t)k0 * Cout];
            b.y = pb[(size_t)k0 * Cout + Cout];
            acc0 = __builtin_amdgcn_wmma_f32_16x16x4_f32(false, a0, false, b, (short)0, acc0, false, false);
            acc1 = __builtin_amdgcn_wmma_f32_16x16x4_f32(false, a1, false, b, (short)0, acc1, false, false);
        }
    }

    if (col < Cout) {
        const float bb = bias ? bias[col] : 0.f;
        const float gg = g    ? g[col]    : 1.f;
        const float tt = bt   ? bt[col]   : 0.f;
        #pragma unroll
        for (int r = 0; r < 8; ++r) {
            // C/D layout: vgpr r -> M=r for lanes 0-15, M=8+r for lanes 16-31
            const int orow = (int)(mt * 32) + half * 8 + r;
            if (orow < M) {
                float y = acc0[r] + bb;
                y = y * gg + tt;
                if (R) y += R[(size_t)orow * Cout + col];
                if (act) y = (y >= 0.f) ? y : NEG_SLOPE * y;
                Y[(size_t)orow * Cout + col] = y;
            }
            const int orow1 = orow + 16;
            if (orow1 < M) {
                float y = acc1[r] + bb;
                y = y * gg + tt;
                if (R) y += R[(size_t)orow1 * Cout + col];
                if (act) y = (y >= 0.f) ? y : NEG_SLOPE * y;
                Y[(size_t)orow1 * Cout + col] = y;
            }
        }
    }
}

// Generic guarded WMMA GEMM (used only for fc0: Cin=3). 16x16 tile per wave.
__device__ __forceinline__ float ldx2(const float* __restrict__ X1, const float* __restrict__ X2,
                                      int C1, int C2, int row, int k) {
    return (k < C1) ? X1[(size_t)row * C1 + k] : X2[(size_t)row * C2 + (k - C1)];
}

__global__ __launch_bounds__(256) void wgemm_generic_kernel(
    const float* __restrict__ X1, int C1,
    const float* __restrict__ X2, int C2,
    const float* __restrict__ W, const float* __restrict__ bias,
    const float* __restrict__ g, const float* __restrict__ bt,
    const float* __restrict__ R, float* __restrict__ Y,
    int M, int Cout, int act)
{
    const int wave = threadIdx.x >> 5;
    const int lane = threadIdx.x & 31;
    const int ntiles = (Cout + 15) >> 4;
    const long long mtiles = (M + 15) >> 4;
    const long long tile = (long long)blockIdx.x * WPB + wave;
    if (tile >= mtiles * ntiles) return;
    const int nt = (int)(tile % ntiles);
    const long long mt = tile / ntiles;
    const int half = lane >> 4;
    const int l16  = lane & 15;
    const int row  = (int)(mt * 16) + l16;
    const int col  = nt * 16 + l16;
    const int Cin  = C1 + C2;
    const int rowc = min(row, M - 1);
    const int colc = min(col, Cout - 1);

    v8f acc = {0.f,0.f,0.f,0.f,0.f,0.f,0.f,0.f};
    for (int k0 = 0; k0 < Cin; k0 += 4) {
        const int ka = k0 + half * 2;
        v2f a, b;
        a.x = (ka     < Cin) ? ldx2(X1, X2, C1, C2, rowc, ka    ) : 0.f;
        a.y = (ka + 1 < Cin) ? ldx2(X1, X2, C1, C2, rowc, ka + 1) : 0.f;
        b.x = (ka     < Cin) ? W[(size_t)(ka    ) * Cout + colc] : 0.f;
        b.y = (ka + 1 < Cin) ? W[(size_t)(ka + 1) * Cout + colc] : 0.f;
        acc = __builtin_amdgcn_wmma_f32_16x16x4_f32(false, a, false, b, (short)0, acc, false, false);
    }

    if (col < Cout) {
        const float bb = bias ? bias[col] : 0.f;
        const float gg = g    ? g[col]    : 1.f;
        const float tt = bt   ? bt[col]   : 0.f;
        #pragma unroll
        for (int r = 0; r < 8; ++r) {
            const int orow = (int)(mt * 16) + half * 8 + r;
            if (orow < M) {
                float y = acc[r] + bb;
                y = y * gg + tt;
                if (R) y += R[(size_t)orow * Cout + col];
                if (act) y = (y >= 0.f) ? y : NEG_SLOPE * y;
                Y[(size_t)orow * Cout + col] = y;
            }
        }
    }
}

// out[b,m,k,:] = feat[b, idx[b,m,k], :]   vectorized over channels (C % 4 == 0)
__global__ void gather4_kernel(const float4* __restrict__ feat, const int* __restrict__ idx,
                               float4* __restrict__ out, int Nsrc, int M, int K, int C4,
                               long long total4)
{
    long long t = (long long)blockIdx.x * blockDim.x + threadIdx.x;
    if (t >= total4) return;
    const int c = (int)(t % C4);
    long long r = t / C4;
    const int k = (int)(r % K); r /= K;
    const int m = (int)(r % M);
    const int b = (int)(r / M);
    const int j = idx[((size_t)b * M + m) * K + k];
    out[t] = feat[((size_t)b * Nsrc + j) * C4 + c];
}

// out[b,m,:] = max_k feat[b, idx[b,m,k], :]   vectorized (C % 4 == 0)
__global__ void maxgather4_kernel(const float4* __restrict__ feat, const int* __restrict__ idx,
                                  float4* __restrict__ out, int Nsrc, int M, int K, int C4,
                                  long long total4)
{
    long long t = (long long)blockIdx.x * blockDim.x + threadIdx.x;
    if (t >= total4) return;
    const int c = (int)(t % C4);
    long long r = t / C4;
    const int m = (int)(r % M);
    const int b = (int)(r / M);
    const int* ip = idx + ((size_t)b * M + m) * K;
    float4 mx = make_float4(-INFINITY, -INFINITY, -INFINITY, -INFINITY);
    for (int k = 0; k < K; ++k) {
        const float4 v = feat[((size_t)b * Nsrc + ip[k]) * C4 + c];
        mx.x = fmaxf(mx.x, v.x); mx.y = fmaxf(mx.y, v.y);
        mx.z = fmaxf(mx.z, v.z); mx.w = fmaxf(mx.w, v.w);
    }
    out[((size_t)b * M + m) * C4 + c] = mx;
}

// Fused rel_pos (10-dim geometric encoding) + bb_mlp1 conv (+BN+lrelu).
__global__ void relpos_conv_kernel(const float* __restrict__ xyz, const int* __restrict__ nidx,
                                   const float* __restrict__ W, const float* __restrict__ bias,
                                   const float* __restrict__ g, const float* __restrict__ bt,
                                   float* __restrict__ out, int N, int K, int Cout,
                                   long long total)
{
    long long t = (long long)blockIdx.x * blockDim.x + threadIdx.x;
    if (t >= total) return;
    const int k = (int)(t % K);
    long long bn = t / K;
    const int n = (int)(bn % N);
    const int b = (int)(bn / N);
    const float* p = xyz + ((size_t)b * N + n) * 3;
    const int j = nidx[((size_t)b * N + n) * K + k];
    const float* q = xyz + ((size_t)b * N + j) * 3;
    const float tx = p[0], ty = p[1], tz = p[2];
    const float nx = q[0], ny = q[1], nz = q[2];
    const float rx = tx - nx, ry = ty - ny, rz = tz - nz;
    const float dis = sqrtf(rx * rx + ry * ry + rz * rz);
    const float v[10] = {dis, rx, ry, rz, tx, ty, tz, nx, ny, nz};
    float* o = out + t * Cout;
    for (int c = 0; c < Cout; ++c) {
        float acc = bias[c];
        #pragma unroll
        for (int jj = 0; jj < 10; ++jj) acc += v[jj] * W[jj * Cout + c];
        acc = acc * g[c] + bt[c];
        o[c] = (acc >= 0.f) ? acc : NEG_SLOPE * acc;
    }
}

// Attention pooling: per (bn,c) softmax over K of S, weighted sum of x=concat(X1,X2).
__global__ void attpool_kernel(const float* __restrict__ X1, int C1,
                               const float* __restrict__ X2, int C2,
                               const float* __restrict__ S, float* __restrict__ out,
                               int K, int d, long long total)
{
    long long t = (long long)blockIdx.x * blockDim.x + threadIdx.x;
    if (t >= total) return;
    const int c = (int)(t % d);
    const long long bn = t / d;
    float s[16];
    float mx = -INFINITY;
    for (int k = 0; k < K; ++k) {
        s[k] = S[((size_t)bn * K + k) * d + c];
        mx = fmaxf(mx, s[k]);
    }
    float sum = 0.f;
    for (int k = 0; k < K; ++k) { s[k] = expf(s[k] - mx); sum += s[k]; }
    const float inv = 1.f / sum;
    float acc = 0.f;
    for (int k = 0; k < K; ++k) {
        const float x = (c < C1) ? X1[((size_t)bn * K + k) * C1 + c]
                                 : X2[((size_t)bn * K + k) * C2 + (c - C1)];
        acc += x * s[k] * inv;
    }
    out[t] = acc;
}

// Y[b,c,n] = X[(b*N+n)*C + c]
__global__ void transpose_out_kernel(const float* __restrict__ X, float* __restrict__ Y,
                                     int N, int C, long long total)
{
    long long t = (long long)blockIdx.x * blockDim.x + threadIdx.x;
    if (t >= total) return;
    const int n = (int)(t % N);
    long long r = t / N;
    const int c = (int)(r % C);
    const int b = (int)(r / C);
    Y[t] = X[((size_t)b * N + n) * C + c];
}

// ---------------------------------------------------------------------------
// Host orchestration
// ---------------------------------------------------------------------------

namespace {

struct Conv { const float* b; const float* bt; const float* g; const float* w; };
struct Att  { const float* fc_b; const float* fc_w; Conv mlp; };
struct Enc  { Att att1, att2; Conv bb1, bb2, mlp1, mlp2, shortcut; };

inline unsigned gup(long long total, int bs) { return (unsigned)((total + bs - 1) / bs); }

inline void gemm(hipStream_t s, const float* X1, int C1, const float* X2, int C2,
                 const Conv& c, bool bn, const float* R, float* Y,
                 long long M, int Cout, int act)
{
    const int ntiles = (Cout + 15) / 16;
    if ((C1 % 4 == 0) && (C2 % 4 == 0)) {
        const long long tiles = ((M + 31) / 32) * ntiles;
        wgemm_fast_kernel<<<gup(tiles, WPB), 256, 0, s>>>(X1, C1, X2, C2, c.w, c.b,
            bn ? c.g : nullptr, bn ? c.bt : nullptr, R, Y, (int)M, Cout, act);
    } else {
        const long long tiles = ((M + 15) / 16) * ntiles;
        wgemm_generic_kernel<<<gup(tiles, WPB), 256, 0, s>>>(X1, C1, X2, C2, c.w, c.b,
            bn ? c.g : nullptr, bn ? c.bt : nullptr, R, Y, (int)M, Cout, act);
    }
}

inline void gather(hipStream_t s, const float* feat, const int* idx, float* out,
                   int B, int Nsrc, int M, int K, int C)
{
    const long long total4 = (long long)B * M * K * (C / 4);
    gather4_kernel<<<gup(total4, 256), 256, 0, s>>>(
        (const float4*)feat, idx, (float4*)out, Nsrc, M, K, C / 4, total4);
}

} // namespace

extern "C" void kernel_launch(void* const* d_in, const int* in_sizes, int n_in,
                              void* d_out, int out_size, void* d_ws, size_t ws_size,
                              hipStream_t stream)
{
    (void)in_sizes; (void)n_in; (void)out_size; (void)ws_size;
    const int B = 2, K = 16;
    const int NP[4] = {65536, 16384, 4096, 1024};
    const int NS[4] = {16384, 4096, 1024, 256};
    const int DO[4] = {16, 64, 128, 256};
    const int DIN[4] = {8, 32, 128, 256};

    // ---- inputs -----------------------------------------------------------
    const float* features = (const float*)d_in[0];
    const float* xyz[4];  const int* nidx[4]; const int* sidx[4]; const int* iidx[4];
    for (int i = 0; i < 4; ++i) {
        xyz[i]  = (const float*)d_in[1 + 4 * i];
        nidx[i] = (const int*)  d_in[2 + 4 * i];
        sidx[i] = (const int*)  d_in[3 + 4 * i];
        iidx[i] = (const int*)  d_in[4 + 4 * i];
    }

    // ---- params: jax pytree flatten order = sorted dict keys at every level
    int pi = 17;
    auto nf = [&]() { return (const float*)d_in[pi++]; };
    auto rdConv = [&]() { Conv c; c.b = nf(); c.bt = nf(); c.g = nf(); c.w = nf(); return c; };
    auto rdAtt  = [&]() { Att a; a.fc_b = nf(); a.fc_w = nf(); a.mlp = rdConv(); return a; };

    Conv dec[5];
    for (int j = 0; j < 5; ++j) dec[j] = rdConv();
    Enc enc[4];
    for (int i = 0; i < 4; ++i) {
        enc[i].att1 = rdAtt(); enc[i].att2 = rdAtt();
        enc[i].bb1 = rdConv(); enc[i].bb2 = rdConv();
        enc[i].mlp1 = rdConv(); enc[i].mlp2 = rdConv(); enc[i].shortcut = rdConv();
    }
    Conv fc0 = rdConv(), fc1 = rdConv(), fc2 = rdConv();
    Conv fc3; fc3.b = nf(); fc3.w = nf(); fc3.g = nullptr; fc3.bt = nullptr;

    // ---- workspace arena (floats) ----------------------------------------
    float* ws = (float*)d_ws;
    size_t off = 0;
    auto A = [&](size_t n) { float* p = ws + off; off += (n + 63) & ~(size_t)63; return p; };
    float* P_FEAT = A((size_t)131072 * 8);
    float* P_ENC0 = A((size_t)131072 * 32);
    float* P_F0   = A((size_t)32768 * 32);
    float* P_F1   = A((size_t)8192 * 128);
    float* P_F2   = A((size_t)2048 * 256);
    float* P_F3   = A((size_t)512 * 512);
    float* P_D0   = A((size_t)512 * 512);
    float* P_D1   = A((size_t)2048 * 256);
    float* P_D2   = A((size_t)8192 * 128);
    float* P_D3   = A((size_t)32768 * 32);
    float* SX = A((size_t)16777216);
    float* SY = A((size_t)16777216);
    float* SS = A((size_t)33554432);
    float* T1 = A((size_t)4194304);
    float* T2 = A((size_t)2097152);
    float* T3 = A((size_t)2097152);
    float* T4 = A((size_t)4194304);
    float* T5 = A((size_t)4194304);

    // ---- fc0 (Cin=3 -> generic guarded path) -----------------------------
    gemm(stream, features, 3, nullptr, 0, fc0, true, nullptr, P_FEAT, (long long)B * NP[0], 8, 1);

    // ---- encoder ----------------------------------------------------------
    const float* fin = P_FEAT;
    float* fsamp[4] = {P_F0, P_F1, P_F2, P_F3};
    for (int i = 0; i < 4; ++i) {
        const int N = NP[i], d = DO[i], d2 = d / 2, din = DIN[i];
        const long long BN = (long long)B * N, BNK = BN * K;
        const Enc& E = enc[i];

        gemm(stream, fin, din, nullptr, 0, E.mlp1, true, nullptr, T1, BN, d2, 1);
        relpos_conv_kernel<<<gup(BNK, 256), 256, 0, stream>>>(
            xyz[i], nidx[i], E.bb1.w, E.bb1.b, E.bb1.g, E.bb1.bt, SX, N, K, d2, BNK);
        gather(stream, T1, nidx[i], SY, B, N, N, K, d2);
        Conv fcc1; fcc1.b = E.att1.fc_b; fcc1.w = E.att1.fc_w; fcc1.g = nullptr; fcc1.bt = nullptr;
        gemm(stream, SY, d2, SX, d2, fcc1, false, nullptr, SS, BNK, d, 0);
        attpool_kernel<<<gup(BN * d, 256), 256, 0, stream>>>(SY, d2, SX, d2, SS, T3, K, d, BN * d);
        gemm(stream, T3, d, nullptr, 0, E.att1.mlp, true, nullptr, T2, BN, d2, 1);
        gemm(stream, SX, d2, nullptr, 0, E.bb2, true, nullptr, SY, BNK, d2, 1);
        gather(stream, T2, nidx[i], SX, B, N, N, K, d2);
        Conv fcc2; fcc2.b = E.att2.fc_b; fcc2.w = E.att2.fc_w; fcc2.g = nullptr; fcc2.bt = nullptr;
        gemm(stream, SX, d2, SY, d2, fcc2, false, nullptr, SS, BNK, d, 0);
        attpool_kernel<<<gup(BN * d, 256), 256, 0, stream>>>(SX, d2, SY, d2, SS, T3, K, d, BN * d);
        gemm(stream, T3, d, nullptr, 0, E.att2.mlp, true, nullptr, T1, BN, d, 1);
        gemm(stream, fin, din, nullptr, 0, E.shortcut, true, nullptr, T4, BN, 2 * d, 0);
        float* fenc = (i == 0) ? P_ENC0 : T5;
        gemm(stream, T1, d, nullptr, 0, E.mlp2, true, T4, fenc, BN, 2 * d, 1);
        const long long tot4 = (long long)B * NS[i] * (2 * d / 4);
        maxgather4_kernel<<<gup(tot4, 256), 256, 0, stream>>>(
            (const float4*)fenc, sidx[i], (float4*)fsamp[i], N, NS[i], K, 2 * d / 4, tot4);
        fin = fsamp[i];
    }

    // ---- decoder ----------------------------------------------------------
    gemm(stream, P_F3, 512, nullptr, 0, dec[0], true, nullptr, P_D0, (long long)B * 256, 512, 1);

    struct { const float* skip; int cskip; const float* src; int nsrc; int ndst; int csrc;
             const int* idx; float* dst; int cout; } DJ[4] = {
        {P_F2, 256, P_D0, 256, 1024, 512, iidx[3], P_D1, 256},
        {P_F1, 128, P_D1, 1024, 4096, 256, iidx[2], P_D2, 128},
        {P_F0, 32, P_D2, 4096, 16384, 128, iidx[1], P_D3, 32},
        {P_ENC0, 32, P_D3, 16384, 65536, 32, iidx[0], SY, 32},
    };
    for (int j = 0; j < 4; ++j) {
        gather(stream, DJ[j].src, DJ[j].idx, SX, B, DJ[j].nsrc, DJ[j].ndst, 1, DJ[j].csrc);
        gemm(stream, DJ[j].skip, DJ[j].cskip, SX, DJ[j].csrc, dec[j + 1], true, nullptr,
             DJ[j].dst, (long long)B * DJ[j].ndst, DJ[j].cout, 1);
    }

    // ---- head: fc1 -> fc2 -> fc3 -> transpose ----------------------------
    const long long BN0 = (long long)B * NP[0];
    gemm(stream, SY, 32, nullptr, 0, fc1, true, nullptr, SX, BN0, 64, 1);
    gemm(stream, SX, 64, nullptr, 0, fc2, true, nullptr, SS, BN0, 32, 1);
    gemm(stream, SS, 32, nullptr, 0, fc3, false, nullptr, T1, BN0, 8, 0);
    const long long tot = BN0 * 8;
    transpose_out_kernel<<<gup(tot, 256), 256, 0, stream>>>(T1, (float*)d_out, NP[0], 8, tot);
}